// AttnDecoderRNN_34205119545959
// MI455X (gfx1250) — compile-verified
//
#include <hip/hip_runtime.h>
#include <math.h>

// Problem constants (match reference)
#define H    2048
#define V    50257
#define S    2048
#define K3   6144   // 3*H
#define KCAT 4096   // 2*H

typedef float v2f __attribute__((ext_vector_type(2)));
typedef float v8f __attribute__((ext_vector_type(8)));

// ---------------------------------------------------------------------------
// x = emb[word]  (gather one 2048-float row)
__global__ void gather_row(const float* __restrict__ emb, const int* __restrict__ idx,
                           float* __restrict__ x) {
  int t = blockIdx.x * blockDim.x + threadIdx.x;
  if (t < H) x[t] = emb[(size_t)idx[0] * H + t];
}

// ---------------------------------------------------------------------------
// Row-major GEMV: y[row] = dot(W[row,:], x) + b[row] + *sadd
// One wave per row, lanes stride by float4 -> fully coalesced 512B/instr.
__global__ void gemv_rows(const float* __restrict__ W, const float* __restrict__ x,
                          const float* __restrict__ b, const float* __restrict__ sadd,
                          float* __restrict__ y, int R, int K) {
  int wave = threadIdx.x >> 5, lane = threadIdx.x & 31;
  int row = blockIdx.x * (blockDim.x >> 5) + wave;
  if (row >= R) return;
  const float* wr = W + (size_t)row * K;
  float s = 0.f;
  for (int k = lane * 4; k < K; k += 128) {
    float4 w4 = *(const float4*)(wr + k);
    float4 x4 = *(const float4*)(x + k);
    s += w4.x * x4.x + w4.y * x4.y;
    s += w4.z * x4.z + w4.w * x4.w;
  }
#pragma unroll
  for (int off = 16; off; off >>= 1) s += __shfl_xor(s, off, 32);
  if (lane == 0) {
    if (b)    s += b[row];
    if (sadd) s += sadd[0];
    y[row] = s;
  }
}

// ---------------------------------------------------------------------------
// GRU gate fusion: h[j] from gi/gh (3H each) and hprev
__global__ void gru_gate(const float* __restrict__ gi, const float* __restrict__ gh,
                         const float* __restrict__ hprev, float* __restrict__ h) {
  int j = blockIdx.x * blockDim.x + threadIdx.x;
  if (j >= H) return;
  float r = 1.f / (1.f + expf(-(gi[j]         + gh[j])));
  float z = 1.f / (1.f + expf(-(gi[H + j]     + gh[H + j])));
  float n = tanhf(gi[2 * H + j] + r * gh[2 * H + j]);
  h[j] = (1.f - z) * n + z * hprev[j];
}

// ---------------------------------------------------------------------------
// Single-workgroup dot product (fixed-order -> deterministic)
__global__ void dotk(const float* __restrict__ a, const float* __restrict__ b,
                     int n, float* __restrict__ outv) {
  __shared__ float red[256];
  int t = threadIdx.x;
  float s = 0.f;
  for (int i = t; i < n; i += 256) s += a[i] * b[i];
  red[t] = s; __syncthreads();
  for (int o = 128; o; o >>= 1) { if (t < o) red[t] += red[t + o]; __syncthreads(); }
  if (t == 0) outv[0] = red[0];
}

// ---------------------------------------------------------------------------
// Column GEMV partials: part[by][k] = sum_{i in chunk by} s[i]*M[i*K+k]
// Coalesced across k; deterministic two-stage reduction (no float atomics).
__global__ void gemv_cols_part(const float* __restrict__ M, const float* __restrict__ s,
                               float* __restrict__ part, int Ni, int K, int ichunk) {
  int k = blockIdx.x * blockDim.x + threadIdx.x;
  if (k >= K) return;
  int i0 = blockIdx.y * ichunk;
  int i1 = min(i0 + ichunk, Ni);
  float acc = 0.f;
  for (int i = i0; i < i1; ++i) acc += s[i] * M[(size_t)i * K + k];
  part[(size_t)blockIdx.y * K + k] = acc;
}

__global__ void reduce_part(const float* __restrict__ part, float* __restrict__ y,
                            int C, int K) {
  int k = blockIdx.x * blockDim.x + threadIdx.x;
  if (k >= K) return;
  float a = 0.f;
  for (int c = 0; c < C; ++c) a += part[(size_t)c * K + k];
  y[k] = a;
}

// ---------------------------------------------------------------------------
// Softmax over S=2048, single workgroup
__global__ void softmax2048(const float* __restrict__ e, float* __restrict__ w) {
  __shared__ float red[256];
  int t = threadIdx.x;
  float m = -1e30f;
  for (int i = t; i < S; i += 256) m = fmaxf(m, e[i]);
  red[t] = m; __syncthreads();
  for (int o = 128; o; o >>= 1) { if (t < o) red[t] = fmaxf(red[t], red[t + o]); __syncthreads(); }
  float M = red[0]; __syncthreads();
  float s = 0.f;
  for (int i = t; i < S; i += 256) s += expf(e[i] - M);
  red[t] = s; __syncthreads();
  for (int o = 128; o; o >>= 1) { if (t < o) red[t] += red[t + o]; __syncthreads(); }
  float inv = 1.f / red[0];
  for (int i = t; i < S; i += 256) w[i] = expf(e[i] - M) * inv;
}

// ---------------------------------------------------------------------------
// Dominant op: logits[m] = dot(w_out[m,:], concat(h1,ctx)) + b_out[m]
// WMMA GEMV with V_WMMA_F32_16X16X4_F32:
//  - 128 threads (4 waves) per WG, 64 output rows per WG (16 rows per wave)
//  - W tiles staged coalesced into LDS (stride 68 floats -> bank-conflict-free
//    strided reads in the 16x4 A layout, float4-aligned stores)
//  - B matrix = input vector broadcast across all 16 columns, so every column
//    of the 16x16 accumulator carries the GEMV result for its row.
// A layout (16x4 f32): VGPR0 = K0 (lanes 0-15) / K2 (lanes 16-31); VGPR1 = K1/K3.
// B layout (4x16 f32): VGPR0 = row K0 (lanes 0-15) / row K2 (lanes 16-31); VGPR1 = K1/K3.
// D layout (16x16 f32): VGPR r = M=r (lanes 0-15) / M=r+8 (lanes 16-31).
__global__ void wout_wmma(const float* __restrict__ Wout, const float* __restrict__ h1,
                          const float* __restrict__ ctx, const float* __restrict__ bout,
                          float* __restrict__ logits) {
  __shared__ float xs[KCAT];        // 16 KB: concat(h1, context)
  __shared__ float tile[64 * 68];   // 17 KB: 64 rows x 64 k, stride 68

  int t    = threadIdx.x;           // 0..127
  int lane = t & 31, wave = t >> 5;
  int mrow = lane & 15, half = lane >> 4;
  int arow = wave * 16 + mrow;      // row within 64-row tile for A reads

  for (int i = t; i < H; i += 128) { xs[i] = h1[i]; xs[H + i] = ctx[i]; }
  __syncthreads();

  int rb_wg = blockIdx.x * 64;
  int lrow  = t >> 1;               // 0..63 : cooperative-load row
  int khalf = (t & 1) * 32;         // which 32-float half of the 64-wide k tile

  v8f acc = {0.f, 0.f, 0.f, 0.f, 0.f, 0.f, 0.f, 0.f};

  for (int k0 = 0; k0 < KCAT; k0 += 64) {
    // ---- coalesced stage of W[rb_wg:rb_wg+64, k0:k0+64] into LDS ----
    int m = rb_wg + lrow;
    const float* src = Wout + (size_t)m * KCAT + k0 + khalf;
#pragma unroll
    for (int i = 0; i < 8; ++i) {
      float4 w4 = (m < V) ? *(const float4*)(src + i * 4)
                          : make_float4(0.f, 0.f, 0.f, 0.f);
      *(float4*)&tile[lrow * 68 + khalf + i * 4] = w4;
    }
    __syncthreads();

    // ---- 16 WMMAs per 64-wide k tile ----
#pragma unroll
    for (int kk = 0; kk < 64; kk += 4) {
      v2f a, b;
      int ao = arow * 68 + kk + half * 2;
      a[0] = tile[ao];
      a[1] = tile[ao + 1];
      int bo = k0 + kk + half * 2;
      b[0] = xs[bo];
      b[1] = xs[bo + 1];
      acc = __builtin_amdgcn_wmma_f32_16x16x4_f32(
          /*neg_a=*/false, a, /*neg_b=*/false, b,
          /*c_mod=*/(short)0, acc, /*reuse_a=*/false, /*reuse_b=*/false);
    }
    __syncthreads();
  }

  // Extract: lane 0 holds rows rb..rb+7 in acc[0..7]; lane 16 holds rb+8..rb+15.
  if ((lane & 15) == 0) {
    int mbase = rb_wg + wave * 16 + half * 8;
#pragma unroll
    for (int r = 0; r < 8; ++r) {
      int m2 = mbase + r;
      if (m2 < V) logits[m2] = acc[r] + bout[m2];
    }
  }
}

// ---------------------------------------------------------------------------
// log_softmax over V=50257, single 1024-thread workgroup (L2-resident data)
__global__ void log_softmax_v(const float* __restrict__ l, float* __restrict__ out) {
  __shared__ float red[1024];
  int t = threadIdx.x;
  float m = -1e30f;
  for (int i = t; i < V; i += 1024) m = fmaxf(m, l[i]);
  red[t] = m; __syncthreads();
  for (int o = 512; o; o >>= 1) { if (t < o) red[t] = fmaxf(red[t], red[t + o]); __syncthreads(); }
  float M = red[0]; __syncthreads();
  float s = 0.f;
  for (int i = t; i < V; i += 1024) s += expf(l[i] - M);
  red[t] = s; __syncthreads();
  for (int o = 512; o; o >>= 1) { if (t < o) red[t] += red[t + o]; __syncthreads(); }
  float L = M + logf(red[0]);
  for (int i = t; i < V; i += 1024) out[i] = l[i] - L;
}

// ---------------------------------------------------------------------------
extern "C" void kernel_launch(void* const* d_in, const int* in_sizes, int n_in,
                              void* d_out, int out_size, void* d_ws, size_t ws_size,
                              hipStream_t stream) {
  const int*   word  = (const int*)  d_in[0];
  const float* lh    = (const float*)d_in[1];   // (2,1,H)
  const float* enc   = (const float*)d_in[2];   // (S,1,H) row-major
  const float* emb   = (const float*)d_in[3];
  const float* w_ih0 = (const float*)d_in[4];
  const float* w_hh0 = (const float*)d_in[5];
  const float* b_ih0 = (const float*)d_in[6];
  const float* b_hh0 = (const float*)d_in[7];
  const float* w_ih1 = (const float*)d_in[8];
  const float* w_hh1 = (const float*)d_in[9];
  const float* b_ih1 = (const float*)d_in[10];
  const float* b_hh1 = (const float*)d_in[11];
  const float* wa    = (const float*)d_in[12];
  const float* ba    = (const float*)d_in[13];
  const float* w_out = (const float*)d_in[14];
  const float* b_out = (const float*)d_in[15];

  float* out = (float*)d_out;   // [log_softmax(V) | h0(H) | h1(H) | attn(S)]
  float* ws  = (float*)d_ws;

  // workspace layout (floats)
  float* x0 = ws;               // 2048  emb row
  float* gi = ws + 2048;        // 6144
  float* gh = ws + 8192;        // 6144
  float* vv = ws + 14336;       // 2048  h1 @ wa
  float* cs = ws + 16384;       // 16    scalar c = h1 . ba
  float* en = ws + 16400;       // 2048  energies
  float* cx = ws + 18448;       // 2048  context
  float* pp = ws + 20496;       // 16*2048 column-GEMV partials
  float* lg = ws + 53264;       // V     logits

  float* h0 = out + V;
  float* h1 = out + V + H;
  float* at = out + V + 2 * H;

  // GRU layer 0
  gather_row<<<8, 256, 0, stream>>>(emb, word, x0);
  gemv_rows<<<768, 256, 0, stream>>>(w_ih0, x0, b_ih0, nullptr, gi, K3, H);
  gemv_rows<<<768, 256, 0, stream>>>(w_hh0, lh,  b_hh0, nullptr, gh, K3, H);
  gru_gate<<<8, 256, 0, stream>>>(gi, gh, lh, h0);
  // GRU layer 1
  gemv_rows<<<768, 256, 0, stream>>>(w_ih1, h0,     b_ih1, nullptr, gi, K3, H);
  gemv_rows<<<768, 256, 0, stream>>>(w_hh1, lh + H, b_hh1, nullptr, gh, K3, H);
  gru_gate<<<8, 256, 0, stream>>>(gi, gh, lh + H, h1);

  // Attention, restructured: energies[s] = (h1@wa) . enc[s] + h1.ba
  dotk<<<1, 256, 0, stream>>>(h1, ba, H, cs);
  gemv_cols_part<<<dim3(8, 16), 256, 0, stream>>>(wa, h1, pp, H, H, 128);
  reduce_part<<<8, 256, 0, stream>>>(pp, vv, 16, H);
  gemv_rows<<<256, 256, 0, stream>>>(enc, vv, nullptr, cs, en, S, H);
  softmax2048<<<1, 256, 0, stream>>>(en, at);
  // context = attn @ enc
  gemv_cols_part<<<dim3(8, 16), 256, 0, stream>>>(enc, at, pp, S, H, 128);
  reduce_part<<<8, 256, 0, stream>>>(pp, cx, 16, H);

  // logits via fp32 WMMA GEMV (823 MB -> the bandwidth-critical op)
  wout_wmma<<<(V + 63) / 64, 128, 0, stream>>>(w_out, h1, cx, b_out, lg);
  log_softmax_v<<<1, 1024, 0, stream>>>(lg, out);
}